// UnitaryLayer_27075473834255
// MI455X (gfx1250) — compile-verified
//
#include <hip/hip_runtime.h>
#include <math.h>

// ---------------- types ----------------
typedef float v2f __attribute__((ext_vector_type(2)));
typedef float v4f __attribute__((ext_vector_type(4)));
typedef float v8f __attribute__((ext_vector_type(8)));

#define UNITS   1024
#define IN_DIM  256
#define OUT_DIM 10
#define BATCH   64
#define SEQ     512
#define KTOT    (IN_DIM + 2*UNITS)      // 2304 augmented K
#define NKT     (KTOT/4)                // 576 k-tiles of 4
#define NNT     (UNITS/16)              // 64 n-tiles
#define AS      2308                    // LDS A row stride (floats), padded for bank spread
#define TWO_PI  6.28318530717958647692f

// ---------------- device scratch (avoid dependence on ws_size) ----------------
__device__ float2 g_s1c[UNITS];                 // c1 * s1[k]
__device__ float2 g_s2c[UNITS];                 // c2 * s2[k]
__device__ float2 g_Amat[UNITS*UNITS];          // D2 @ (R1 @ fft(D1))[:,perm]   (8 MB)
__device__ float2 g_t3[UNITS*UNITS];            // ifft(...)                      (8 MB)
__device__ v4f    g_Bpk[NNT*NKT*32];            // WMMA-fragment-packed {Br0,Br1,Bi0,Bi1} (18 MB)

__device__ __forceinline__ float2 cmulf(float2 a, float2 b) {
    return make_float2(a.x*b.x - a.y*b.y, a.x*b.y + a.y*b.x);
}
__device__ __forceinline__ float2 cisf(float a) {
    float s, c; sincosf(a, &s, &c); return make_float2(c, s);
}

// ---------------- K1: s1c[k] = (2/||r1||^2) * sum_l conj(r1[l]) e^{i d1[l]} e^{-2pi i lk/N}
__global__ void k_s1(const float* __restrict__ d1, const float* __restrict__ r1) {
    int k = blockIdx.x, tid = threadIdx.x;
    __shared__ float red[3][256];
    float sr = 0.f, si = 0.f, n2 = 0.f;
    for (int l = tid; l < UNITS; l += 256) {
        float rr = r1[2*l], ri = r1[2*l+1];
        n2 += rr*rr + ri*ri;
        float ang = d1[l] - TWO_PI * (float)((l*k) & (UNITS-1)) * (1.0f/UNITS);
        float s, c; sincosf(ang, &s, &c);
        sr += rr*c + ri*s;        // conj(r1)*cis(ang)
        si += rr*s - ri*c;
    }
    red[0][tid] = sr; red[1][tid] = si; red[2][tid] = n2;
    __syncthreads();
    for (int o = 128; o > 0; o >>= 1) {
        if (tid < o) { red[0][tid]+=red[0][tid+o]; red[1][tid]+=red[1][tid+o]; red[2][tid]+=red[2][tid+o]; }
        __syncthreads();
    }
    if (tid == 0) {
        float c1 = 2.0f / red[2][0];
        g_s1c[k] = make_float2(c1*red[0][0], c1*red[1][0]);
    }
}

// ---------------- K2: Amat[j][m] = e^{i d2[j]} * ( e^{i d1[j]} e^{-2pi i j*perm[m]/N} - r1[j]*s1c[perm[m]] )
__global__ void k_amat(const float* __restrict__ d1, const float* __restrict__ d2,
                       const float* __restrict__ r1, const int* __restrict__ perm) {
    int idx = blockIdx.x*256 + threadIdx.x;
    int j = idx >> 10, m = idx & (UNITS-1);
    int p = perm[m];
    float th = TWO_PI * (float)((j*p) & (UNITS-1)) * (1.0f/UNITS);
    float2 t1 = cisf(d1[j] + d2[j] - th);
    float2 rj = make_float2(r1[2*j], r1[2*j+1]);
    float2 t2 = cmulf(cisf(d2[j]), cmulf(g_s1c[p], rj));
    g_Amat[idx] = make_float2(t1.x - t2.x, t1.y - t2.y);
}

// ---------------- K3: t3[j][k] = (1/N) sum_m Amat[j][m] e^{+2pi i mk/N}   (row-wise IDFT, exact twiddles)
__global__ void k_idft() {
    __shared__ float2 tab[UNITS];
    int tid = threadIdx.x;
    for (int i = tid; i < UNITS; i += 256) tab[i] = cisf(TWO_PI * (float)i * (1.0f/UNITS));
    __syncthreads();
    int j = blockIdx.x >> 2;
    int k = ((blockIdx.x & 3) << 8) + tid;
    const float2* row = g_Amat + j*UNITS;
    float ar = 0.f, ai = 0.f;
    for (int m = 0; m < UNITS; ++m) {
        float2 w = tab[(m*k) & (UNITS-1)];
        float2 a = row[m];
        ar += a.x*w.x - a.y*w.y;
        ai += a.x*w.y + a.y*w.x;
    }
    g_t3[j*UNITS + k] = make_float2(ar*(1.0f/UNITS), ai*(1.0f/UNITS));
}

// ---------------- K4: s2c[k] = (2/||r2||^2) * sum_l conj(r2[l]) t3[l][k]
__global__ void k_s2(const float* __restrict__ r2) {
    int k = blockIdx.x, tid = threadIdx.x;
    __shared__ float red[3][256];
    float sr = 0.f, si = 0.f, n2 = 0.f;
    for (int l = tid; l < UNITS; l += 256) {
        float rr = r2[2*l], ri = r2[2*l+1];
        n2 += rr*rr + ri*ri;
        float2 t = g_t3[l*UNITS + k];
        sr += rr*t.x + ri*t.y;
        si += rr*t.y - ri*t.x;
    }
    red[0][tid] = sr; red[1][tid] = si; red[2][tid] = n2;
    __syncthreads();
    for (int o = 128; o > 0; o >>= 1) {
        if (tid < o) { red[0][tid]+=red[0][tid+o]; red[1][tid]+=red[1][tid+o]; red[2][tid]+=red[2][tid+o]; }
        __syncthreads();
    }
    if (tid == 0) {
        float c2 = 2.0f / red[2][0];
        g_s2c[k] = make_float2(c2*red[0][0], c2*red[1][0]);
    }
}

// ---------------- K5: finalize W and pack augmented B = [V ; W ; rot(W)] into WMMA fragment order
// Br rows: [Vr ; Wr ; -Wi]   Bi rows: [Vi ; Wi ; Wr]
__global__ void k_pack(const float* __restrict__ V, const float* __restrict__ d3,
                       const float* __restrict__ r2) {
    int idx = blockIdx.x*256 + threadIdx.x;    // idx = r*1024 + n
    int r = idx >> 10, n = idx & (UNITS-1);
    float br, bi;
    if (r < IN_DIM) {
        br = V[2*idx]; bi = V[2*idx + 1];      // V row-major (256,1024) interleaved complex
    } else {
        int u = (r < IN_DIM + UNITS) ? (r - IN_DIM) : (r - IN_DIM - UNITS);
        float2 t  = g_t3[u*UNITS + n];
        float2 ru = make_float2(r2[2*u], r2[2*u+1]);
        float2 sc = g_s2c[n];
        float2 rs = cmulf(sc, ru);
        float2 w  = cmulf(cisf(d3[u]), make_float2(t.x - rs.x, t.y - rs.y));
        if (r < IN_DIM + UNITS) { br = w.x;  bi = w.y; }   // Wr / Wi
        else                    { br = -w.y; bi = w.x; }   // -Wi / Wr
    }
    // fragment mapping: VGPR s, lane l  <->  row = kt*4 + 2*(l>>4) + s, col = ntile*16 + (l&15)
    int kt = r >> 2, rr = r & 3, s = rr & 1, lhi = rr >> 1;
    int lane  = lhi*16 + (n & 15);
    int ntile = n >> 4;
    float* bp = (float*)g_Bpk;
    size_t base = ((size_t)(ntile*NKT + kt)*32 + lane) * 4;
    bp[base + s]     = br;
    bp[base + 2 + s] = bi;
}

// ---------------- K6: persistent scan kernel (4 blocks x 512 threads = 16 waves each)
__global__ void __launch_bounds__(512)
k_scan(const float* __restrict__ inputs, const float* __restrict__ bias,
       const float* __restrict__ x0, const float* __restrict__ U,
       const float* __restrict__ bias_out, float* __restrict__ out) {
    extern __shared__ float sm[];
    float* A     = sm;                  // 16 rows x AS floats: [x_t(256) | Xr(1024) | Xi(1024)]
    float* biasl = sm + 16*AS;          // 1024
    float* ybuf  = biasl + UNITS;       // 160

    const int tid   = threadIdx.x;
    const int chunk = blockIdx.x;       // 4 chunks of 16 batch rows
    const int wave  = tid >> 5;
    const int lane  = tid & 31;
    const int mrow  = lane & 15;
    const int khalf = (lane >> 4) * 2;  // 0 or 2

    for (int u = tid; u < UNITS; u += 512) biasl[u] = bias[u];
    for (int e = tid; e < 16*UNITS; e += 512) {
        int m = e >> 10, u = e & (UNITS-1);
        A[m*AS + IN_DIM + u]         = x0[2*u];
        A[m*AS + IN_DIM + UNITS + u] = x0[2*u + 1];
    }

    const float* inbase = inputs + (size_t)chunk * 16 * SEQ * IN_DIM;
    const float* Arow = A + mrow*AS;

    for (int t = 0; t < SEQ; ++t) {
        // stage x_t (16 x 256) into A cols [0,256)
        for (int e = tid; e < 16*IN_DIM; e += 512) {
            int m = e >> 8, d = e & (IN_DIM-1);
            A[m*AS + d] = inbase[((size_t)m*SEQ + t)*IN_DIM + d];
        }
        __syncthreads();

        v8f accR[4], accI[4];
        #pragma unroll
        for (int i = 0; i < 4; ++i) { accR[i] = (v8f)(0.0f); accI[i] = (v8f)(0.0f); }

        #pragma unroll 1
        for (int kt = 0; kt < NKT; ++kt) {
            v2f a = *(const v2f*)(Arow + kt*4 + khalf);      // ds_load_b64 A-fragment
            #pragma unroll
            for (int i = 0; i < 4; ++i) {
                int nt = wave + (i << 4);
                v4f b = g_Bpk[(size_t)(nt*NKT + kt)*32 + lane];  // global_load_b128 Br+Bi fragments
                v2f brv = { b[0], b[1] };
                v2f biv = { b[2], b[3] };
                accR[i] = __builtin_amdgcn_wmma_f32_16x16x4_f32(
                    false, a, false, brv, (short)0, accR[i], false, false);
                accI[i] = __builtin_amdgcn_wmma_f32_16x16x4_f32(
                    false, a, false, biv, (short)0, accI[i], false, false);
            }
        }
        __syncthreads();   // all reads of old state done

        // modReLU in registers, write new state back to LDS
        #pragma unroll
        for (int i = 0; i < 4; ++i) {
            int nt = wave + (i << 4);
            int ng = nt*16 + (lane & 15);
            float bb = biasl[ng];
            #pragma unroll
            for (int r = 0; r < 8; ++r) {
                int m = r + khalf*4;                 // lanes 16-31 -> M = r+8
                float zr = accR[i][r], zi = accI[i][r];
                float nrm = sqrtf(zr*zr + zi*zi);
                float M = fmaxf(nrm + bb, 0.0f) / nrm;
                A[m*AS + IN_DIM + ng]         = M*zr;
                A[m*AS + IN_DIM + UNITS + ng] = M*zi;
            }
        }
        // next iteration stages disjoint cols then barriers before any read
    }
    __syncthreads();

    // readout: y = [Re(x)|Im(x)] @ U + bias_out  (x_flat[m][j] == A[m][256+j], j in [0,2048))
    if (tid < 16*OUT_DIM) {
        int m = tid / OUT_DIM, o = tid % OUT_DIM;
        float acc = bias_out[o];
        const float* xf = A + m*AS + IN_DIM;
        for (int j = 0; j < 2*UNITS; ++j) acc += xf[j] * U[j*OUT_DIM + o];
        ybuf[m*OUT_DIM + o] = acc;
    }
    __syncthreads();
    if (tid < 16) {
        float mx = -1e30f;
        for (int o = 0; o < OUT_DIM; ++o) mx = fmaxf(mx, ybuf[tid*OUT_DIM + o]);
        float e[OUT_DIM], s = 0.f;
        for (int o = 0; o < OUT_DIM; ++o) { e[o] = expf(ybuf[tid*OUT_DIM + o] - mx); s += e[o]; }
        for (int o = 0; o < OUT_DIM; ++o) out[(chunk*16 + tid)*OUT_DIM + o] = e[o] / s;
    }
}

// ---------------- host ----------------
extern "C" void kernel_launch(void* const* d_in, const int* in_sizes, int n_in,
                              void* d_out, int out_size, void* d_ws, size_t ws_size,
                              hipStream_t stream) {
    const float* inputs   = (const float*)d_in[0];
    const float* d1       = (const float*)d_in[1];
    const float* d2       = (const float*)d_in[2];
    const float* d3       = (const float*)d_in[3];
    const float* r1       = (const float*)d_in[4];   // complex64 interleaved
    const float* r2       = (const float*)d_in[5];   // complex64 interleaved
    const float* V        = (const float*)d_in[6];   // complex64 interleaved (256,1024)
    const float* U        = (const float*)d_in[7];   // (2048,10)
    const float* bias     = (const float*)d_in[8];
    const float* bias_out = (const float*)d_in[9];
    const float* x0       = (const float*)d_in[10];  // complex64 interleaved
    // perm = d_in[11]
    const int* perm       = (const int*)d_in[11];
    float* out = (float*)d_out;

    k_s1  <<<UNITS, 256, 0, stream>>>(d1, r1);
    k_amat<<<UNITS*UNITS/256, 256, 0, stream>>>(d1, d2, r1, perm);
    k_idft<<<UNITS*4, 256, 0, stream>>>();
    k_s2  <<<UNITS, 256, 0, stream>>>(r2);
    k_pack<<<KTOT*UNITS/256, 256, 0, stream>>>(V, d3, r2);

    size_t smem = (size_t)(16*AS + UNITS + 192) * sizeof(float);  // ~152 KB (<320 KB WGP LDS)
    k_scan<<<4, 512, smem, stream>>>(inputs, bias, x0, U, bias_out, out);
}